// ChebconvProcessor_28999619182969
// MI455X (gfx1250) — compile-verified
//
#include <hip/hip_runtime.h>
#include <hip/hip_bf16.h>

// ---------------------------------------------------------------------------
// ClusterGCN x3 (+LeakyReLU, residual, train-mode BatchNorm) for MI455X.
// N=100000 nodes, C=32 channels, E=3.2M edges.
// Edge scatter is L2-atomic bound; GEMMs use exact-fp32 v_wmma_f32_16x16x4_f32.
// ---------------------------------------------------------------------------

#define N_NODES 100000
#define N_EDGES 3200000
#define CH      32
#define EPS     1e-5f
#define SLOPE   0.01f

typedef __attribute__((ext_vector_type(2))) float v2f;
typedef __attribute__((ext_vector_type(8))) float v8f;

// ---------------- degree (layer-invariant) ----------------
__global__ void k_deg_init(float* __restrict__ deg) {
    int i = blockIdx.x * blockDim.x + threadIdx.x;
    if (i < N_NODES) deg[i] = 1.0f;                 // self loop
}

__global__ void k_deg_count(const int* __restrict__ ei, float* __restrict__ deg) {
    int e = blockIdx.x * blockDim.x + threadIdx.x;
    if (e < N_EDGES) atomicAdd(&deg[ei[N_EDGES + e]], 1.0f);   // col = ei[E..2E)
}

__global__ void k_deg_invert(float* __restrict__ deg) {
    int i = blockIdx.x * blockDim.x + threadIdx.x;
    if (i < N_NODES) deg[i] = 1.0f / deg[i];        // deg >= 1 always -> clip is a no-op
}

// ---------------- aggregation ----------------
__global__ void k_copy(const float* __restrict__ src, float* __restrict__ dst) {
    int i = blockIdx.x * blockDim.x + threadIdx.x;
    if (i < N_NODES * CH) dst[i] = src[i];
}

// one thread per (edge, channel): wave32 = one full 128B row, coalesced atomics
__global__ __launch_bounds__(256)
void k_scatter(const int* __restrict__ ei, const float* __restrict__ x,
               float* __restrict__ agg) {
    int idx = blockIdx.x * blockDim.x + threadIdx.x;          // < E*32 = 102.4M
    if (idx >= N_EDGES * CH) return;
    int e  = idx >> 5;
    int ch = idx & 31;
    int r  = ei[e];
    int c  = ei[N_EDGES + e];
    atomicAdd(&agg[(size_t)c * CH + ch], x[(size_t)r * CH + ch]);
}

// ---------------- fused GCN GEMM epilogue (WMMA fp32) ----------------
// y = maybeLeaky( deginv*agg @ Wout + bout + x @ Wroot ) + x
// One wave handles a 16-row tile (16x32 output = two 16x16 C/D fragments).
template <bool LEAKY>
__global__ __launch_bounds__(256)
void k_gcn_gemm(const float* __restrict__ agg, const float* __restrict__ xin,
                const float* __restrict__ deginv,
                const float* __restrict__ Wout, const float* __restrict__ bout,
                const float* __restrict__ Wroot,
                float* __restrict__ y) {
    const int lane = threadIdx.x & 31;
    const int wave = threadIdx.x >> 5;
    const int tile = blockIdx.x * 8 + wave;
    if (tile >= N_NODES / 16) return;               // whole-wave uniform exit, EXEC stays all-1s

    const int m  = lane & 15;                       // A row within tile / B,C column
    const int hi = lane >> 4;                       // lane half selects K pair (A/B layout)
    const int arow = tile * 16 + m;
    const float di = deginv[arow];
    const float* aggRow = agg + (size_t)arow * CH;
    const float* xRow   = xin + (size_t)arow * CH;
    const int koff = hi * 2;

    v8f acc0 = {};                                  // cols 0..15
    v8f acc1 = {};                                  // cols 16..31
#pragma unroll
    for (int k0 = 0; k0 < CH; k0 += 4) {
        const int kb = k0 + koff;                   // this lane's two K indices: kb, kb+1
        // A fragments (16x4 fp32): lane l holds (M=l&15, K=kb),(M,kb+1)
        v2f aAgg = { aggRow[kb] * di, aggRow[kb + 1] * di };
        v2f aX   = { xRow[kb],        xRow[kb + 1] };
        // B fragments (4x16 fp32): lane l holds (K=kb, N=l&15),(K=kb+1, N)
        v2f bo0 = { Wout [kb * CH + m],       Wout [(kb + 1) * CH + m] };
        v2f bo1 = { Wout [kb * CH + 16 + m],  Wout [(kb + 1) * CH + 16 + m] };
        v2f br0 = { Wroot[kb * CH + m],       Wroot[(kb + 1) * CH + m] };
        v2f br1 = { Wroot[kb * CH + 16 + m],  Wroot[(kb + 1) * CH + 16 + m] };
        // both GEMMs chained into the same accumulators (exact fp32)
        acc0 = __builtin_amdgcn_wmma_f32_16x16x4_f32(false, aAgg, false, bo0, (short)0, acc0, false, false);
        acc1 = __builtin_amdgcn_wmma_f32_16x16x4_f32(false, aAgg, false, bo1, (short)0, acc1, false, false);
        acc0 = __builtin_amdgcn_wmma_f32_16x16x4_f32(false, aX,   false, br0, (short)0, acc0, false, false);
        acc1 = __builtin_amdgcn_wmma_f32_16x16x4_f32(false, aX,   false, br1, (short)0, acc1, false, false);
    }

    const float bias0 = bout[m];
    const float bias1 = bout[16 + m];
    // C/D layout: VGPR v, lane l -> row = v + 8*(l>>4), col = l&15 (+n0)
#pragma unroll
    for (int v = 0; v < 8; ++v) {
        const int r = tile * 16 + v + 8 * hi;
        float g0 = acc0[v] + bias0;
        float g1 = acc1[v] + bias1;
        if (LEAKY) {
            g0 = g0 >= 0.f ? g0 : SLOPE * g0;
            g1 = g1 >= 0.f ? g1 : SLOPE * g1;
        }
        y[(size_t)r * CH + m]      = g0 + xin[(size_t)r * CH + m];
        y[(size_t)r * CH + 16 + m] = g1 + xin[(size_t)r * CH + 16 + m];
    }
}

// ---------------- BatchNorm (training-mode, biased var) ----------------
__global__ void k_stats_zero(float* __restrict__ stats) {
    if (threadIdx.x < 2 * CH) stats[threadIdx.x] = 0.f;
}

__global__ __launch_bounds__(256)
void k_bn_stats(const float* __restrict__ y, float* __restrict__ stats) {
    __shared__ float ssum[CH], ssq[CH];
    if (threadIdx.x < CH) { ssum[threadIdx.x] = 0.f; ssq[threadIdx.x] = 0.f; }
    __syncthreads();
    const int ch  = threadIdx.x & 31;
    const int rpb = blockDim.x >> 5;                // rows visited per block per pass
    float s = 0.f, q = 0.f;
    for (int r = blockIdx.x * rpb + (threadIdx.x >> 5); r < N_NODES; r += gridDim.x * rpb) {
        float v = y[(size_t)r * CH + ch];           // coalesced 128B per wave
        s += v; q += v * v;
    }
    atomicAdd(&ssum[ch], s);
    atomicAdd(&ssq[ch], q);
    __syncthreads();
    if (threadIdx.x < CH) {
        atomicAdd(&stats[threadIdx.x], ssum[threadIdx.x]);
        atomicAdd(&stats[CH + threadIdx.x], ssq[threadIdx.x]);
    }
}

__global__ void k_bn_norm(const float* __restrict__ y, const float* __restrict__ stats,
                          const float* __restrict__ g, const float* __restrict__ be,
                          float* __restrict__ out) {
    int i = blockIdx.x * blockDim.x + threadIdx.x;
    if (i >= N_NODES * CH) return;
    int ch = i & 31;
    const float invN = 1.0f / (float)N_NODES;
    float mean = stats[ch] * invN;
    float var  = stats[CH + ch] * invN - mean * mean;   // biased var (matches reference)
    float inv  = rsqrtf(var + EPS);
    out[i] = (y[i] - mean) * inv * g[ch] + be[ch];
}

// ---------------------------------------------------------------------------
extern "C" void kernel_launch(void* const* d_in, const int* in_sizes, int n_in,
                              void* d_out, int out_size, void* d_ws, size_t ws_size,
                              hipStream_t stream) {
    (void)in_sizes; (void)n_in; (void)out_size; (void)ws_size;

    const float* x0  = (const float*)d_in[0];
    const int*   ei  = (const int*)d_in[1];        // int32 (JAX x64 disabled); [row(E) | col(E)]
    // d_in[2] = edge_attr: unused by the reference
    const float* W1o = (const float*)d_in[3],  *b1o = (const float*)d_in[4];
    const float* W1r = (const float*)d_in[5],  *g1  = (const float*)d_in[6],  *be1 = (const float*)d_in[7];
    const float* W2o = (const float*)d_in[8],  *b2o = (const float*)d_in[9];
    const float* W2r = (const float*)d_in[10], *g2  = (const float*)d_in[11], *be2 = (const float*)d_in[12];
    const float* W3o = (const float*)d_in[13], *b3o = (const float*)d_in[14];
    const float* W3r = (const float*)d_in[15], *g3  = (const float*)d_in[16], *be3 = (const float*)d_in[17];
    float* out = (float*)d_out;

    // workspace: deg(N) | agg(N*C) | Y(N*C) | X(N*C) | stats(64)  ~= 39 MB
    float* deg   = (float*)d_ws;
    float* agg   = deg + N_NODES;
    float* Y     = agg + (size_t)N_NODES * CH;
    float* X     = Y + (size_t)N_NODES * CH;
    float* stats = X + (size_t)N_NODES * CH;

    const int TB  = 256;
    const int gN  = (N_NODES + TB - 1) / TB;
    const int gNC = (N_NODES * CH + TB - 1) / TB;
    const int gE  = (N_EDGES + TB - 1) / TB;
    const int gEC = (N_EDGES * CH + TB - 1) / TB;      // 400000 blocks
    const int gT  = (N_NODES / 16 + 7) / 8;            // 8 tiles (waves) per block

    // degree normalization (shared by all layers)
    k_deg_init  <<<gN, TB, 0, stream>>>(deg);
    k_deg_count <<<gE, TB, 0, stream>>>(ei, deg);
    k_deg_invert<<<gN, TB, 0, stream>>>(deg);

    // ---- layer 1: x0 -> Y (pre-BN) -> X ----
    k_copy   <<<gNC, TB, 0, stream>>>(x0, agg);
    k_scatter<<<gEC, TB, 0, stream>>>(ei, x0, agg);
    k_gcn_gemm<true><<<gT, TB, 0, stream>>>(agg, x0, deg, W1o, b1o, W1r, Y);
    k_stats_zero<<<1, 64, 0, stream>>>(stats);
    k_bn_stats<<<1024, TB, 0, stream>>>(Y, stats);
    k_bn_norm <<<gNC, TB, 0, stream>>>(Y, stats, g1, be1, X);

    // ---- layer 2: X -> Y (pre-BN) -> Y (normalize in place) ----
    k_copy   <<<gNC, TB, 0, stream>>>(X, agg);
    k_scatter<<<gEC, TB, 0, stream>>>(ei, X, agg);
    k_gcn_gemm<true><<<gT, TB, 0, stream>>>(agg, X, deg, W2o, b2o, W2r, Y);
    k_stats_zero<<<1, 64, 0, stream>>>(stats);
    k_bn_stats<<<1024, TB, 0, stream>>>(Y, stats);
    k_bn_norm <<<gNC, TB, 0, stream>>>(Y, stats, g2, be2, Y);

    // ---- layer 3 (no leaky): Y -> out (pre-BN) -> out (in place) ----
    k_copy   <<<gNC, TB, 0, stream>>>(Y, agg);
    k_scatter<<<gEC, TB, 0, stream>>>(ei, Y, agg);
    k_gcn_gemm<false><<<gT, TB, 0, stream>>>(agg, Y, deg, W3o, b3o, W3r, out);
    k_stats_zero<<<1, 64, 0, stream>>>(stats);
    k_bn_stats<<<1024, TB, 0, stream>>>(out, stats);
    k_bn_norm <<<gNC, TB, 0, stream>>>(out, stats, g3, be3, out);
}